// SpatialAttention_65051574665858
// MI455X (gfx1250) — compile-verified
//
#include <hip/hip_runtime.h>

typedef float v2f __attribute__((ext_vector_type(2)));
typedef float v4f __attribute__((ext_vector_type(4)));
typedef float v8f __attribute__((ext_vector_type(8)));

static constexpr int kB  = 16;
static constexpr int kC  = 256;
static constexpr int kW  = 128;
static constexpr int kHW = 128 * 128;   // 16384

// ---------------------------------------------------------------------------
// Kernel 1: channel mean + max over C=256. Pure HBM-bandwidth bound (268 MB).
// Each thread owns 4 consecutive pixels (one b128 load per channel), NT hint
// since x has zero reuse. 65536 threads, unroll 16 -> ~16MB loads in flight.
// Output layout: s[b][0][hw] = mean, s[b][1][hw] = max  (matches concat order).
// ---------------------------------------------------------------------------
__global__ __launch_bounds__(256) void reduce_mean_max(
    const float* __restrict__ x, float* __restrict__ s) {
  int t   = blockIdx.x * 256 + threadIdx.x;   // 0..65535
  int b   = t >> 12;                          // 4096 float4-groups per image
  int hw4 = (t & 4095) << 2;

  const v4f* p = (const v4f*)(x + (size_t)b * kC * kHW + hw4);
  v4f sum = {0.f, 0.f, 0.f, 0.f};
  float ninf = -__builtin_inff();
  v4f mx = {ninf, ninf, ninf, ninf};

#pragma unroll 16
  for (int c = 0; c < kC; ++c) {
    v4f v = __builtin_nontemporal_load(p + c * (kHW / 4));  // 64KB stride
    sum += v;
    mx[0] = fmaxf(mx[0], v[0]);
    mx[1] = fmaxf(mx[1], v[1]);
    mx[2] = fmaxf(mx[2], v[2]);
    mx[3] = fmaxf(mx[3], v[3]);
  }
  v4f avg = sum * (1.0f / 256.0f);

  v4f* sp = (v4f*)(s + (size_t)b * 2 * kHW);
  sp[hw4 >> 2]              = avg;   // plane 0: mean
  sp[(kHW >> 2) + (hw4 >> 2)] = mx;  // plane 1: max
}

// ---------------------------------------------------------------------------
// Kernel 2: 7x7 conv (2->1 ch) + sigmoid via V_WMMA_F32_16X16X4_F32.
// Per 16x16 output tile: D[j,i] = sum_k A[j,k] * B[k,i]
//   k = (c, yy, dx), yy in [0,22), dx in [0,7)  ->  K = 308 = 77*4
//   A[j,k] = s[c, y0+yy-3, x0+j+dx-3]   (LDS tile, zero padded)
//   B[k,i] = W[c, yy-i, dx] if 0<=yy-i<7 else 0   (Toeplitz weights)
// One wave32 per tile; EXEC all-1s as required by WMMA.
// ---------------------------------------------------------------------------
__global__ __launch_bounds__(32) void conv7x7_wmma(
    const float* __restrict__ s, const float* __restrict__ w,
    float* __restrict__ out) {
  __shared__ float tile[2 * 22 * 24];   // 22 rows x 22 cols (stride 24), 2 ch
  __shared__ float wl[98];              // [c][dy][dx]

  int lane = threadIdx.x;               // 0..31
  int bid  = blockIdx.x;                // 0..1023
  int b    = bid >> 6;
  int tl   = bid & 63;
  int y0   = (tl >> 3) << 4;
  int x0   = (tl & 7) << 4;

  const float* sb = s + (size_t)b * 2 * kHW;

  for (int i = lane; i < 98; i += 32) wl[i] = w[i];

  for (int pos = lane; pos < 2 * 484; pos += 32) {
    int c  = (pos >= 484);
    int r  = pos - c * 484;
    int yy = r / 22;
    int xx = r - yy * 22;
    int gy = y0 + yy - 3;
    int gx = x0 + xx - 3;
    float v = 0.0f;
    if ((unsigned)gy < 128u && (unsigned)gx < 128u)
      v = sb[c * kHW + gy * kW + gx];
    tile[c * 528 + yy * 24 + xx] = v;
  }
  __syncthreads();

  int half = lane >> 4;     // which K-pair this lane holds (A: K 0,1 vs 2,3)
  int mn   = lane & 15;     // A row j (lanes' M) and B column i (lanes' N)

  // Incremental decomposition of kk = 4*step + 2*half into (c0, yy0, dx0).
  int c0 = 0, yy0 = 0, dx0 = half * 2;   // 0 or 2, both < 7

  v8f acc = {0.f, 0.f, 0.f, 0.f, 0.f, 0.f, 0.f, 0.f};

  for (int step = 0; step < 77; ++step) {
    // element 1 indices = element 0 + 1 (wrap dx->yy->c)
    int dx1 = dx0 + 1, yy1 = yy0, c1 = c0;
    if (dx1 == 7) { dx1 = 0; ++yy1; if (yy1 == 22) { yy1 = 0; ++c1; } }

    v2f a, bb;
    a.x = tile[c0 * 528 + yy0 * 24 + mn + dx0];
    a.y = tile[c1 * 528 + yy1 * 24 + mn + dx1];

    int dyA = yy0 - mn;
    int dyB = yy1 - mn;
    int iA = dyA < 0 ? 0 : (dyA > 6 ? 6 : dyA);
    int iB = dyB < 0 ? 0 : (dyB > 6 ? 6 : dyB);
    float wA = wl[c0 * 49 + iA * 7 + dx0];
    float wB = wl[c1 * 49 + iB * 7 + dx1];
    bb.x = ((unsigned)dyA < 7u) ? wA : 0.0f;
    bb.y = ((unsigned)dyB < 7u) ? wB : 0.0f;

    acc = __builtin_amdgcn_wmma_f32_16x16x4_f32(
        false, a, false, bb, (short)0, acc, false, false);

    // advance element-0 k by 4
    dx0 += 4;
    if (dx0 >= 7) { dx0 -= 7; ++yy0; if (yy0 == 22) { yy0 = 0; ++c0; } }
  }

  // D layout: lane<16 -> M=v,   N=lane;  lane>=16 -> M=v+8, N=lane-16
  // M = x-offset j, N = y-offset i.
  int iy = mn;
  int jb = half * 8;
  float* ob = out + (size_t)b * kHW + (y0 + iy) * kW + (x0 + jb);
#pragma unroll
  for (int v = 0; v < 8; ++v) {
    float val = acc[v];
    ob[v] = 1.0f / (1.0f + __expf(-val));
  }
}

// ---------------------------------------------------------------------------
extern "C" void kernel_launch(void* const* d_in, const int* in_sizes, int n_in,
                              void* d_out, int out_size, void* d_ws, size_t ws_size,
                              hipStream_t stream) {
  const float* x = (const float*)d_in[0];   // [16,256,128,128]
  const float* w = (const float*)d_in[1];   // [1,2,7,7]
  float* out = (float*)d_out;               // [16,1,128,128]
  float* s   = (float*)d_ws;                // [16,2,128,128] = 2 MB scratch

  reduce_mean_max<<<256, 256, 0, stream>>>(x, s);
  conv7x7_wmma<<<kB * 64, 32, 0, stream>>>(s, w, out);
}